// GraphMultisetTransformer_57475252355502
// MI455X (gfx1250) — compile-verified
//
#include <hip/hip_runtime.h>
#include <math.h>

// ---------------------------------------------------------------------------
// Graph Multiset Transformer forward, fp32, WMMA f32 16x16x4 path (gfx1250)
// ---------------------------------------------------------------------------

typedef float v2f __attribute__((ext_vector_type(2)));
typedef float v8f __attribute__((ext_vector_type(8)));

#define TOTAL_N 65536
#define EDGES   1048576
#define HDIM    128
#define BATCH   256
#define NPG     256
#define SEEDS   75
#define SEEDP   80
#define ZROWS   (BATCH * SEEDP)      // 20480
#define ATTN_CHUNK 64

static __device__ __forceinline__ v8f wmma_f32(v2f a, v2f b, v8f c) {
  // D = A(16x4,f32) * B(4x16,f32) + C(16x16,f32)
  return __builtin_amdgcn_wmma_f32_16x16x4_f32(false, a, false, b, (short)0, c,
                                               false, false);
}

// ---------------------------------------------------------------------------
// Generic GEMM: C[M,128] = A[M,128] @ W[128,128] (+bias) (+relu-residual)
// grid.x = M/16 (one 16-row M tile per block); 8 waves cover N=128.
// pad_mod/pad_valid: rows with (row % pad_mod) >= pad_valid are forced to 0.
// ---------------------------------------------------------------------------
__global__ __launch_bounds__(256) void gemm128_kernel(
    const float* __restrict__ A, const float* __restrict__ W,
    const float* __restrict__ bias, const float* __restrict__ R,
    float* __restrict__ C, int relu_res, int pad_mod, int pad_valid) {
  const int m0   = blockIdx.x * 16;
  const int wave = threadIdx.x >> 5;
  const int lane = threadIdx.x & 31;
  const int half = lane >> 4;
  const int l16  = lane & 15;
  const int n0   = wave * 16;

  const float* Arow = A + (size_t)(m0 + l16) * HDIM;
  v8f acc = {0.f, 0.f, 0.f, 0.f, 0.f, 0.f, 0.f, 0.f};
  for (int k0 = 0; k0 < HDIM; k0 += 4) {
    const int ka = k0 + half * 2;
    v2f a; a.x = Arow[ka];                  a.y = Arow[ka + 1];
    v2f b; b.x = W[ka * HDIM + n0 + l16];   b.y = W[(ka + 1) * HDIM + n0 + l16];
    acc = wmma_f32(a, b, acc);
  }
  const int col = n0 + l16;
  const float bv = bias ? bias[col] : 0.0f;
  for (int r = 0; r < 8; ++r) {
    const int row = m0 + r + half * 8;
    float v = acc[r] + bv;
    if (relu_res) v = R[(size_t)row * HDIM + col] + fmaxf(v, 0.0f);
    if (pad_mod && (row % pad_mod) >= pad_valid) v = 0.0f;
    C[(size_t)row * HDIM + col] = v;
  }
}

// ---------------------------------------------------------------------------
// Degree / GCN helpers
// ---------------------------------------------------------------------------
__global__ void deg_init_kernel(float* deg, int n) {
  int i = blockIdx.x * 256 + threadIdx.x;
  if (i < n) deg[i] = 1.0f;  // self loop
}
__global__ void deg_scatter_kernel(const int* __restrict__ dst, float* deg, int e) {
  int i = blockIdx.x * 256 + threadIdx.x;
  if (i < e) atomicAdd(&deg[dst[i]], 1.0f);
}
__global__ void deg_rsqrt_kernel(float* deg, int n) {
  int i = blockIdx.x * 256 + threadIdx.x;
  if (i < n) deg[i] = rsqrtf(deg[i]);  // deg >= 1 always
}
// out[i,:] = bias + dinv[i]^2 * xw[i,:]   (self-loop term + bias)
__global__ void gcn_init_kernel(const float* __restrict__ xw,
                                const float* __restrict__ dinv,
                                const float* __restrict__ bias,
                                float* __restrict__ out) {
  int idx = blockIdx.x * 256 + threadIdx.x;  // TOTAL_N*128 = 8.4M < 2^31
  if (idx >= TOTAL_N * HDIM) return;
  int i = idx >> 7, d = idx & 127;
  float di = dinv[i];
  out[idx] = bias[d] + di * di * xw[idx];
}
// out[dst,:] += dinv[src]*dinv[dst] * xw[src,:]
__global__ void gcn_scatter_kernel(const float* __restrict__ xw,
                                   const int* __restrict__ src,
                                   const int* __restrict__ dst,
                                   const float* __restrict__ dinv,
                                   float* __restrict__ out) {
  long idx = (long)blockIdx.x * 256 + threadIdx.x;  // EDGES*128
  if (idx >= (long)EDGES * HDIM) return;
  int e = (int)(idx >> 7), d = (int)(idx & 127);
  int s = src[e], t = dst[e];
  float nrm = dinv[s] * dinv[t];
  atomicAdd(out + (size_t)t * HDIM + d, nrm * xw[(size_t)s * HDIM + d]);
}

// ---------------------------------------------------------------------------
// Pad-copy: dst[rows_pad,128] <- src[rows_valid,128], zero pad rows.
// ---------------------------------------------------------------------------
__global__ void pad_rows_kernel(const float* __restrict__ src, float* __restrict__ dst,
                                int rows_valid, int rows_pad) {
  int idx = blockIdx.x * 256 + threadIdx.x;
  if (idx >= rows_pad * HDIM) return;
  int r = idx >> 7;
  dst[idx] = (r < rows_valid) ? src[idx] : 0.0f;
}

// ---------------------------------------------------------------------------
// Attention (MAB core): out[b,q,:] = Qh[q,:] + sum_k softmax_q(QK^T/s)[q,k] V[k,:]
//   - softmax over the QUERY axis (75 valid queries), per key column
//   - Q padded to 80 rows (pad rows are zero); pad A rows set to 0
//   - K/V laid out [BATCH, kv_rows, 128]; key columns beyond valid rows are
//     zero rows -> contribute nothing (A_col * 0)
// grid.x = BATCH*4 (batch,head); block = 320 threads = 10 waves.
// wave w owns output tile (mi=w>>1 in 0..4, ni=w&1 in 0..1).
// ---------------------------------------------------------------------------
__global__ __launch_bounds__(320) void attn_kernel(
    const float* __restrict__ Qh, int q_batch_stride_rows,  // 0 => shared Q
    const float* __restrict__ K, const float* __restrict__ V,
    int kv_rows, int nk_pad, float* __restrict__ out, float scale) {
  __shared__ float sS[SEEDP * ATTN_CHUNK];  // 80 x 64 scores chunk (20 KB)

  const int b    = blockIdx.x >> 2;
  const int h    = blockIdx.x & 3;
  const int wave = threadIdx.x >> 5;
  const int lane = threadIdx.x & 31;
  const int half = lane >> 4;
  const int l16  = lane & 15;
  const int hc   = h * 32;

  const float* Qb = Qh + (size_t)b * q_batch_stride_rows * HDIM;
  const float* Kb = K + (size_t)b * kv_rows * HDIM;
  const float* Vb = V + (size_t)b * kv_rows * HDIM;

  const int mi = wave >> 1, ni = wave & 1;
  v8f acc = {0.f, 0.f, 0.f, 0.f, 0.f, 0.f, 0.f, 0.f};

  for (int c0 = 0; c0 < nk_pad; c0 += ATTN_CHUNK) {
    const int cw = (nk_pad - c0 < ATTN_CHUNK) ? (nk_pad - c0) : ATTN_CHUNK;
    const int ntiles = cw >> 4;

    // --- scores: S[q, c] = (Q[q,:32] . K[c0+c,:32]) * scale  via WMMA ------
    const int tot = 5 * ntiles;
    for (int t = wave; t < tot; t += 10) {
      const int q0  = (t / ntiles) * 16;
      const int n0l = (t % ntiles) * 16;
      const float* Aq = Qb + (size_t)(q0 + l16) * HDIM + hc;
      const float* Kc = Kb + (size_t)(c0 + n0l + l16) * HDIM + hc;
      v8f sacc = {0.f, 0.f, 0.f, 0.f, 0.f, 0.f, 0.f, 0.f};
      for (int k0 = 0; k0 < 32; k0 += 4) {
        const int ka = k0 + half * 2;
        v2f a;  a.x = Aq[ka];  a.y = Aq[ka + 1];
        v2f bb; bb.x = Kc[ka]; bb.y = Kc[ka + 1];   // B = K^T tile
        sacc = wmma_f32(a, bb, sacc);
      }
      for (int r = 0; r < 8; ++r)
        sS[(q0 + r + half * 8) * ATTN_CHUNK + n0l + l16] = sacc[r] * scale;
    }
    __syncthreads();

    // --- softmax over q (valid 0..74) per key column -----------------------
    for (int col = threadIdx.x; col < cw; col += 320) {
      float mx = -1e30f;
      for (int q = 0; q < SEEDS; ++q) mx = fmaxf(mx, sS[q * ATTN_CHUNK + col]);
      float sum = 0.f;
      for (int q = 0; q < SEEDS; ++q) {
        float e = __expf(sS[q * ATTN_CHUNK + col] - mx);
        sS[q * ATTN_CHUNK + col] = e;
        sum += e;
      }
      float inv = 1.0f / sum;
      for (int q = 0; q < SEEDS; ++q) sS[q * ATTN_CHUNK + col] *= inv;
      for (int q = SEEDS; q < SEEDP; ++q) sS[q * ATTN_CHUNK + col] = 0.0f;
    }
    __syncthreads();

    // --- context: acc[q,d] += A_chunk[q,:] @ V_chunk[:,d]  via WMMA --------
    for (int kk = 0; kk < cw; kk += 4) {
      const int ka = kk + half * 2;
      v2f a;
      a.x = sS[(mi * 16 + l16) * ATTN_CHUNK + ka];
      a.y = sS[(mi * 16 + l16) * ATTN_CHUNK + ka + 1];
      const float* Vc = Vb + (size_t)(c0 + ka) * HDIM + hc + ni * 16 + l16;
      v2f bb; bb.x = Vc[0]; bb.y = Vc[HDIM];
      acc = wmma_f32(a, bb, acc);
    }
    __syncthreads();
  }

  // residual + store: out[b*80+q, hc+ni*16+l16]
  for (int r = 0; r < 8; ++r) {
    const int q   = mi * 16 + r + half * 8;
    const int col = hc + ni * 16 + l16;
    out[((size_t)b * SEEDP + q) * HDIM + col] = Qb[(size_t)q * HDIM + col] + acc[r];
  }
}

// ---------------------------------------------------------------------------
// LayerNorm over 128 channels; one row per wave, 8 rows per 256-thread block.
// ---------------------------------------------------------------------------
__global__ __launch_bounds__(256) void layernorm_kernel(
    const float* __restrict__ X, const float* __restrict__ g,
    const float* __restrict__ bt, float* __restrict__ Y, int M, int pad_mod,
    int pad_valid) {
  const int row = blockIdx.x * 8 + (threadIdx.x >> 5);
  if (row >= M) return;
  const int lane = threadIdx.x & 31;
  const float* x = X + (size_t)row * HDIM;
  float v[4], s = 0.f, s2 = 0.f;
  for (int i = 0; i < 4; ++i) {
    v[i] = x[lane * 4 + i];
    s += v[i];
    s2 += v[i] * v[i];
  }
  for (int off = 16; off >= 1; off >>= 1) {
    s  += __shfl_xor(s, off, 32);
    s2 += __shfl_xor(s2, off, 32);
  }
  const float mu   = s * (1.0f / HDIM);
  const float var  = s2 * (1.0f / HDIM) - mu * mu;
  const float rinv = rsqrtf(var + 1e-5f);
  const bool  padz = pad_mod && (row % pad_mod) >= pad_valid;
  for (int i = 0; i < 4; ++i) {
    const int c = lane * 4 + i;
    float y = (v[i] - mu) * rinv * g[c] + bt[c];
    Y[(size_t)row * HDIM + c] = padz ? 0.0f : y;
  }
}

// ---------------------------------------------------------------------------
// PMA2 collapse: softmax over q=1 => A==1 => t[b,:] = q2row + sum_{k<75} v2[b,k,:]
// ---------------------------------------------------------------------------
__global__ void pma2_reduce_kernel(const float* __restrict__ q2row,
                                   const float* __restrict__ v2,
                                   float* __restrict__ t) {
  const int b = blockIdx.x, d = threadIdx.x;  // 128 threads
  float s = q2row[d];
  const float* vb = v2 + (size_t)b * SEEDP * HDIM + d;
  for (int k = 0; k < SEEDS; ++k) s += vb[(size_t)k * HDIM];
  t[(size_t)b * HDIM + d] = s;
}

// ---------------------------------------------------------------------------
// Host orchestration
// ---------------------------------------------------------------------------
static inline void gemm128(const float* A, const float* W, const float* bias,
                           const float* R, float* C, int M, int relu_res,
                           int pad_mod, int pad_valid, hipStream_t s) {
  gemm128_kernel<<<M / 16, 256, 0, s>>>(A, W, bias, R, C, relu_res, pad_mod,
                                        pad_valid);
}
static inline void layernorm(const float* X, const float* g, const float* b,
                             float* Y, int M, int pad_mod, int pad_valid,
                             hipStream_t s) {
  layernorm_kernel<<<(M + 7) / 8, 256, 0, s>>>(X, g, b, Y, M, pad_mod, pad_valid);
}

extern "C" void kernel_launch(void* const* d_in, const int* in_sizes, int n_in,
                              void* d_out, int out_size, void* d_ws,
                              size_t ws_size, hipStream_t stream) {
  (void)in_sizes; (void)n_in; (void)out_size; (void)ws_size;
  const float SCALE = 0.08838834764831845f;  // 1/sqrt(128)

  // ---- inputs (setup_inputs dict order, depth-first) ----
  const float* X    = (const float*)d_in[0];
  const int*   EIX  = (const int*)d_in[2];
  const int*   esrc = EIX;
  const int*   edst = EIX + EDGES;
  const float *Wli = (const float*)d_in[3],  *bli = (const float*)d_in[4];
  // pma1
  const float *Wq1 = (const float*)d_in[5],  *bq1 = (const float*)d_in[6];
  const float *Wk1 = (const float*)d_in[7],  *bk1 = (const float*)d_in[8];
  const float *Wv1 = (const float*)d_in[9],  *bv1 = (const float*)d_in[10];
  const float *Wo1 = (const float*)d_in[11], *bo1 = (const float*)d_in[12];
  const float *g01 = (const float*)d_in[13], *c01 = (const float*)d_in[14];
  const float *g11 = (const float*)d_in[15], *c11 = (const float*)d_in[16];
  const float *S1  = (const float*)d_in[17];
  // sab
  const float *Wqs = (const float*)d_in[18], *bqs = (const float*)d_in[19];
  const float *Wks = (const float*)d_in[20], *bks = (const float*)d_in[21];
  const float *Wvs = (const float*)d_in[22], *bvs = (const float*)d_in[23];
  const float *Wos = (const float*)d_in[24], *bos = (const float*)d_in[25];
  const float *g0s = (const float*)d_in[26], *c0s = (const float*)d_in[27];
  const float *g1s = (const float*)d_in[28], *c1s = (const float*)d_in[29];
  // pma2 (its "k" linear is dead: q=1 -> softmax over q == 1)
  const float *Wq2 = (const float*)d_in[30], *bq2 = (const float*)d_in[31];
  const float *Wv2 = (const float*)d_in[34], *bv2 = (const float*)d_in[35];
  const float *Wo2 = (const float*)d_in[36], *bo2 = (const float*)d_in[37];
  const float *g02 = (const float*)d_in[38], *c02 = (const float*)d_in[39];
  const float *g12 = (const float*)d_in[40], *c12 = (const float*)d_in[41];
  const float *S2  = (const float*)d_in[42];
  const float *Wlo = (const float*)d_in[43], *blo = (const float*)d_in[44];

  // ---- workspace layout (floats) ----
  float* ws = (float*)d_ws;
  const size_t NF = (size_t)TOTAL_N * HDIM;   // 8,388,608
  const size_t Z  = (size_t)ZROWS * HDIM;     // 2,621,440
  float* h    = ws;
  float* xw   = h + NF;
  float* Kg   = xw + NF;
  float* Vg   = Kg + NF;
  float* dinv = Vg + NF;                       // TOTAL_N
  float* S1p  = dinv + TOTAL_N;                // 80*128
  float* Qh1  = S1p + SEEDP * HDIM;            // 80*128
  float* zA   = Qh1 + SEEDP * HDIM;
  float* zB   = zA + Z;
  float* zC   = zB + Z;
  float* zD   = zC + Z;
  float* s2p  = zD + Z;                        // 16*128
  float* q2p  = s2p + 16 * HDIM;               // 16*128
  float* tA   = q2p + 16 * HDIM;               // 256*128
  float* tB   = tA + BATCH * HDIM;             // 256*128

  // 1) h = x @ W_in + b
  gemm128(X, Wli, bli, nullptr, h, TOTAL_N, 0, 0, 0, stream);

  // 2) degree -> dinv  (self loops included)
  deg_init_kernel<<<TOTAL_N / 256, 256, 0, stream>>>(dinv, TOTAL_N);
  deg_scatter_kernel<<<EDGES / 256, 256, 0, stream>>>(edst, dinv, EDGES);
  deg_rsqrt_kernel<<<TOTAL_N / 256, 256, 0, stream>>>(dinv, TOTAL_N);

  // 3) GCN keys:   Kg = D^-1/2 (A+I) D^-1/2 (h Wk1) + bk1
  gemm128(h, Wk1, nullptr, nullptr, xw, TOTAL_N, 0, 0, 0, stream);
  gcn_init_kernel<<<(TOTAL_N * HDIM) / 256, 256, 0, stream>>>(xw, dinv, bk1, Kg);
  gcn_scatter_kernel<<<(EDGES * HDIM) / 256, 256, 0, stream>>>(xw, esrc, edst,
                                                               dinv, Kg);
  // 4) GCN values
  gemm128(h, Wv1, nullptr, nullptr, xw, TOTAL_N, 0, 0, 0, stream);
  gcn_init_kernel<<<(TOTAL_N * HDIM) / 256, 256, 0, stream>>>(xw, dinv, bv1, Vg);
  gcn_scatter_kernel<<<(EDGES * HDIM) / 256, 256, 0, stream>>>(xw, esrc, edst,
                                                               dinv, Vg);

  // 5) PMA1: Qh1 = fc_q(S1) once (shared across batch), padded 75->80
  pad_rows_kernel<<<(SEEDP * HDIM + 255) / 256, 256, 0, stream>>>(S1, S1p, SEEDS,
                                                                  SEEDP);
  gemm128(S1p, Wq1, bq1, nullptr, Qh1, SEEDP, 0, SEEDP, SEEDS, stream);
  //    attention against GCN K/V ([256 rows per graph]; dense-padding is a no-op)
  attn_kernel<<<BATCH * 4, 320, 0, stream>>>(Qh1, 0, Kg, Vg, NPG, NPG, zA, SCALE);
  layernorm(zA, g01, c01, zB, ZROWS, SEEDP, SEEDS, stream);
  gemm128(zB, Wo1, bo1, zB, zC, ZROWS, 1, SEEDP, SEEDS, stream);
  layernorm(zC, g11, c11, zA, ZROWS, SEEDP, SEEDS, stream);   // z1 in zA

  // 6) SAB
  gemm128(zA, Wqs, bqs, nullptr, zB, ZROWS, 0, SEEDP, SEEDS, stream);  // Q
  gemm128(zA, Wks, bks, nullptr, zC, ZROWS, 0, SEEDP, SEEDS, stream);  // K
  gemm128(zA, Wvs, bvs, nullptr, zD, ZROWS, 0, SEEDP, SEEDS, stream);  // V
  attn_kernel<<<BATCH * 4, 320, 0, stream>>>(zB, SEEDP, zC, zD, SEEDP, SEEDP,
                                             zA, SCALE);
  layernorm(zA, g0s, c0s, zB, ZROWS, SEEDP, SEEDS, stream);
  gemm128(zB, Wos, bos, zB, zC, ZROWS, 1, SEEDP, SEEDS, stream);
  layernorm(zC, g1s, c1s, zA, ZROWS, SEEDP, SEEDS, stream);   // z2 in zA

  // 7) PMA2: A==1 (softmax over q=1) -> t = fc_q(S2) + sum_k v(z2)
  gemm128(zA, Wv2, bv2, nullptr, zB, ZROWS, 0, SEEDP, SEEDS, stream);
  pad_rows_kernel<<<(16 * HDIM + 255) / 256, 256, 0, stream>>>(S2, s2p, 1, 16);
  gemm128(s2p, Wq2, bq2, nullptr, q2p, 16, 0, 16, 1, stream);
  pma2_reduce_kernel<<<BATCH, HDIM, 0, stream>>>(q2p, zB, tA);
  layernorm(tA, g02, c02, tB, BATCH, 0, 0, stream);
  gemm128(tB, Wo2, bo2, tB, tA, BATCH, 1, 0, 0, stream);
  layernorm(tA, g12, c12, tB, BATCH, 0, 0, stream);

  // 8) output
  gemm128(tB, Wlo, blo, nullptr, (float*)d_out, BATCH, 0, 0, 0, stream);
}